// HanningLoss_19335942767116
// MI455X (gfx1250) — compile-verified
//
#include <hip/hip_runtime.h>
#include <math.h>

#define B_   64
#define H_   512
#define W_   512
#define K_   33
#define HW_  (H_ * W_)          // 262144 = 2^18
#define CHUNKS 64               // chunks per sample
#define ITERS  4                // float4 loads per thread in main kernel
#define TPB    256
#define PPB    (TPB * 4 * ITERS)    // 4096 pixels per block
#define NBLK   (B_ * CHUNKS)        // 4096 blocks in main pass

typedef __attribute__((ext_vector_type(2))) float v2f;
typedef __attribute__((ext_vector_type(8))) float v8f;

// Wave32 full reduction via V_WMMA_F32_16X16X4_F32 with an all-ones B matrix.
// A layout (16x4 f32): VGPR0 lanes0-15 = A[m,0], lanes16-31 = A[m,2];
// putting each lane's value in a.x gives rowsum[m] = x[m] + x[m+16].
// D[m,n] = rowsum[m]; summing the 8 D VGPRs gives sum(rows 0-7) in lanes<16
// and sum(rows 8-15) in lanes>=16; one xor-16 shuffle completes the sum.
__device__ __forceinline__ float wave_reduce(float x) {
  v2f a; a[0] = x;    a[1] = 0.0f;
  v2f b; b[0] = 1.0f; b[1] = 1.0f;
  v8f c = {};
  c = __builtin_amdgcn_wmma_f32_16x16x4_f32(false, a, false, b,
                                            (short)0, c, false, false);
  float t = c[0] + c[1] + c[2] + c[3] + c[4] + c[5] + c[6] + c[7];
  t += __shfl_xor(t, 16, 32);
  return t;
}

__device__ __forceinline__ float block_reduce(float x, float* sred) {
  float ws = wave_reduce(x);
  int wave = threadIdx.x >> 5, lane = threadIdx.x & 31;
  if (lane == 0) sred[wave] = ws;
  __syncthreads();
  float tot = 0.0f;
  if (wave == 0) {
    float v = (lane < (int)(TPB / 32)) ? sred[lane] : 0.0f;
    tot = wave_reduce(v);
  }
  return tot; // valid in wave 0
}

__global__ void hl_init(unsigned* __restrict__ minidx) {
  if (threadIdx.x < B_) minidx[threadIdx.x] = 0xFFFFFFFFu;
}

// Pass 1: min linear index of nonzero target per sample == ymin*W + xmin.
__global__ void hl_bbox(const float4* __restrict__ tgt,
                        unsigned* __restrict__ minidx, int n4) {
  int i = blockIdx.x * blockDim.x + threadIdx.x;
  int stride = gridDim.x * blockDim.x;
  for (; i < n4; i += stride) {
    float4 v = tgt[i];
    int e = i << 2;
    int b = e >> 18;                 // / HW_
    unsigned pix = (unsigned)(e & (HW_ - 1));
    unsigned best = 0xFFFFFFFFu;
    if      (v.x != 0.0f) best = pix;
    else if (v.y != 0.0f) best = pix + 1u;
    else if (v.z != 0.0f) best = pix + 2u;
    else if (v.w != 0.0f) best = pix + 3u;
    if (best != 0xFFFFFFFFu) atomicMin(&minidx[b], best);
  }
}

// Pass 2: weighted BCE partial sums, one partial per block (deterministic).
__global__ void hl_main(const float* __restrict__ pred,
                        const unsigned* __restrict__ minidx,
                        float* __restrict__ partials,
                        float posScale, float negW) {
  const float STEP = 0.19634954084936207f; // 2*pi/32
  int b     = blockIdx.x / CHUNKS;
  int chunk = blockIdx.x % CHUNKS;
  const float4* pv = (const float4*)(pred + (size_t)b * HW_);
  unsigned mi = minidx[b];
  int ymin = (int)(mi >> 9), xmin = (int)(mi & 511u);

  float acc = 0.0f;
#pragma unroll
  for (int it = 0; it < ITERS; ++it) {
    int v4  = chunk * (PPB / 4) + it * TPB + (int)threadIdx.x;
    float4 p4 = pv[v4];
    int pix = v4 << 2;
    int y   = pix >> 9;
    int x0  = pix & 511;
    int ry  = y - ymin;
    bool rowBlk = ((unsigned)ry < 33u);            // target row
    bool rowPos = ((unsigned)(ry - 1) < 31u);      // positive-weight row
    float hy = rowPos ? 0.5f * (1.0f - cosf((float)ry * STEP)) : 0.0f;
    float pe[4] = {p4.x, p4.y, p4.z, p4.w};
#pragma unroll
    for (int j = 0; j < 4; ++j) {
      int rx = x0 + j - xmin;
      bool inBlk = rowBlk && ((unsigned)rx < 33u);
      bool inPos = rowPos && ((unsigned)(rx - 1) < 31u);
      float w = negW;
      if (inPos)
        w = hy * (0.5f * (1.0f - cosf((float)rx * STEP))) * posScale;
      float t = inBlk ? 1.0f : 0.0f;
      float p = pe[j];
      float l = fmaxf(p, 0.0f) - p * t + log1pf(__expf(-fabsf(p)));
      acc += w * l;
    }
  }

  __shared__ float sred[TPB / 32];
  float tot = block_reduce(acc, sred);
  if (threadIdx.x == 0) partials[blockIdx.x] = tot;
}

// Pass 3: deterministic final reduction + divide by batch.
__global__ void hl_final(const float* __restrict__ partials,
                         float* __restrict__ out) {
  float acc = 0.0f;
  for (int i = (int)threadIdx.x; i < NBLK; i += TPB) acc += partials[i];
  __shared__ float sred[TPB / 32];
  float tot = block_reduce(acc, sred);
  if (threadIdx.x == 0) out[0] = tot * (1.0f / (float)B_);
}

extern "C" void kernel_launch(void* const* d_in, const int* in_sizes, int n_in,
                              void* d_out, int out_size, void* d_ws, size_t ws_size,
                              hipStream_t stream) {
  const float*  pred = (const float*)d_in[0];
  const float4* tgt4 = (const float4*)d_in[1];
  float* out = (float*)d_out;

  unsigned* minidx = (unsigned*)d_ws;                       // 64 * 4 B
  float* partials  = (float*)((char*)d_ws + 256);           // 4096 * 4 B

  // Collapsed normalization constants (see analysis):
  //   w_pos = h(ry)*h(rx) / (2 * S^2),  S = sum_t hann(t)  (== 16 analytically)
  //   w_neg = 1 / (2 * (H*W - 31*31))
  float hs = 0.0f;
  for (int t = 0; t < K_; ++t)
    hs += 0.5f * (1.0f - cosf((float)t * 0.19634954084936207f));
  float posScale = 1.0f / (2.0f * hs * hs);
  float negW = 1.0f / (2.0f * (float)(HW_ - 31 * 31));

  int n4 = (B_ * HW_) / 4;
  hl_init<<<1, 64, 0, stream>>>(minidx);
  hl_bbox<<<4096, TPB, 0, stream>>>(tgt4, minidx, n4);
  hl_main<<<NBLK, TPB, 0, stream>>>(pred, minidx, partials, posScale, negW);
  hl_final<<<1, TPB, 0, stream>>>(partials, out);
}